// TropicalLinear_47614007443911
// MI455X (gfx1250) — compile-verified
//
#include <hip/hip_runtime.h>
#include <cstdint>

// Tropical (max-plus) linear layer: y[b,i] = max_j (W[i,j] + x[b,j])
// x: (B, 512) f32 row-major, W: (512, 512) f32 row-major, y: (B, 512) f32.
//
// Compute-bound VALU kernel (matrix pipes cannot do max-plus). CDNA5 paths
// used: global_load_async_to_lds_b32 (ASYNCcnt, per-lane LDS addressing gives
// free k-major transpose), s_wait_asynccnt double buffering, wave32 dual-issue
// friendly 4x4 register tiles fed by ds_load_b128.

namespace {
constexpr int IN_F  = 512;
constexpr int OUT_F = 512;
constexpr int BM    = 64;    // batch rows per workgroup
constexpr int BN    = 64;    // output cols per workgroup
constexpr int BK    = 16;    // k-slab per stage
constexpr int LSTR  = 68;    // padded LDS row stride (floats): 272B, 16B aligned
constexpr int TILE  = BK * LSTR;
constexpr int NTHR  = 256;   // 8 waves
constexpr int KTILES = IN_F / BK;  // 32
}

__device__ __forceinline__ void issue_tile_async(
    const float* __restrict__ x, const float* __restrict__ W,
    int tid, int b_base, int i_base, int k_base,
    uint32_t lds_x, uint32_t lds_w) {
  // x tile: BM x BK dwords (1024) ; W tile: BN x BK dwords (1024).
  // 4 dwords of each per thread. Consecutive lanes read consecutive k within a
  // row (coalesced 64B runs); LDS destination is transposed: [k][row].
#pragma unroll
  for (int j = 0; j < 4; ++j) {
    const int n = tid + j * NTHR;       // 0..1023
    const int r = n >> 4;               // row within tile (batch b or out i)
    const int k = n & (BK - 1);         // k within slab
    const uint32_t gx = (uint32_t)(((b_base + r) * IN_F + (k_base + k)) * 4);
    const uint32_t lx = lds_x + (uint32_t)((k * LSTR + r) * 4);
    asm volatile("global_load_async_to_lds_b32 %0, %1, %2"
                 :: "v"(lx), "v"(gx), "s"(x) : "memory");
    const uint32_t gw = (uint32_t)(((i_base + r) * IN_F + (k_base + k)) * 4);
    const uint32_t lw = lds_w + (uint32_t)((k * LSTR + r) * 4);
    asm volatile("global_load_async_to_lds_b32 %0, %1, %2"
                 :: "v"(lw), "v"(gw), "s"(W) : "memory");
  }
}

__global__ __launch_bounds__(NTHR)
void tropical_linear_kernel(const float* __restrict__ x,
                            const float* __restrict__ W,
                            float* __restrict__ out) {
  __shared__ __align__(16) float lds[2][2][TILE];  // [buf][x|w][k*LSTR + row]

  const int tid = (int)threadIdx.x;
  const int tx  = tid & 15;   // -> 4 output columns
  const int ty  = tid >> 4;   // -> 4 batch rows
  const int b_base = (int)blockIdx.y * BM;
  const int i_base = (int)blockIdx.x * BN;

  float acc[4][4];
#pragma unroll
  for (int m = 0; m < 4; ++m)
#pragma unroll
    for (int n = 0; n < 4; ++n) acc[m][n] = -__builtin_inff();

  // Low 32 bits of a flat shared pointer == LDS byte offset on gfx1250.
  const uint32_t lx0 = (uint32_t)(uintptr_t)&lds[0][0][0];
  const uint32_t lw0 = (uint32_t)(uintptr_t)&lds[0][1][0];
  const uint32_t lx1 = (uint32_t)(uintptr_t)&lds[1][0][0];
  const uint32_t lw1 = (uint32_t)(uintptr_t)&lds[1][1][0];

  issue_tile_async(x, W, tid, b_base, i_base, 0, lx0, lw0);

  for (int t = 0; t < KTILES; ++t) {
    const int cur = t & 1;
    if (t + 1 < KTILES) {
      // Prefetch next slab into the other buffer (WAR-safe: trailing barrier
      // of the previous iteration already fenced all reads of that buffer).
      issue_tile_async(x, W, tid, b_base, i_base, (t + 1) * BK,
                       cur ? lx0 : lx1, cur ? lw0 : lw1);
      // Async loads complete in order: <=8 outstanding means tile t landed.
      asm volatile("s_wait_asynccnt 8" ::: "memory");
    } else {
      asm volatile("s_wait_asynccnt 0" ::: "memory");
    }
    __syncthreads();

    const float* xs = &lds[cur][0][0];
    const float* ws = &lds[cur][1][0];
#pragma unroll
    for (int k = 0; k < BK; ++k) {
      // One ds_load_b128 each; ws addresses sweep all 64 banks, xs broadcasts.
      const float4 xv = *(const float4*)(xs + k * LSTR + ty * 4);
      const float4 wv = *(const float4*)(ws + k * LSTR + tx * 4);
      const float xm[4] = {xv.x, xv.y, xv.z, xv.w};
      const float wn[4] = {wv.x, wv.y, wv.z, wv.w};
#pragma unroll
      for (int m = 0; m < 4; ++m)
#pragma unroll
        for (int n = 0; n < 4; ++n)
          acc[m][n] = __builtin_fmaxf(acc[m][n], xm[m] + wn[n]);
    }
    __syncthreads();
  }

  const int b0 = b_base + ty * 4;
  const int i0 = i_base + tx * 4;
#pragma unroll
  for (int m = 0; m < 4; ++m) {
    float4 v = make_float4(acc[m][0], acc[m][1], acc[m][2], acc[m][3]);
    *(float4*)(out + (size_t)(b0 + m) * OUT_F + i0) = v;  // global_store_b128
  }
}

extern "C" void kernel_launch(void* const* d_in, const int* in_sizes, int n_in,
                              void* d_out, int out_size, void* d_ws, size_t ws_size,
                              hipStream_t stream) {
  const float* x = (const float*)d_in[0];
  const float* W = (const float*)d_in[1];
  float* out = (float*)d_out;
  const int batch = in_sizes[0] / IN_F;       // 2048
  dim3 grid(OUT_F / BN, batch / BM);          // (8, 32) = 256 workgroups
  dim3 block(NTHR);
  hipLaunchKernelGGL(tropical_linear_kernel, grid, block, 0, stream, x, W, out);
}